// DCLayer_ADMM_57758720196916
// MI455X (gfx1250) — compile-verified
//
#include <hip/hip_runtime.h>
#include <hip/hip_bf16.h>

typedef __attribute__((ext_vector_type(2))) float v2f;
typedef __attribute__((ext_vector_type(8))) float v8f;
typedef __attribute__((ext_vector_type(4))) unsigned int v4u;
typedef __attribute__((ext_vector_type(8))) int v8i;
typedef __attribute__((ext_vector_type(4))) int v4i;

constexpr int BATCH = 1024;
constexpr int MDIM  = 64;
constexpr int NDIM  = 8192;
constexpr int KC    = 256;   // K-chunk staged in LDS for big-K GEMMs
constexpr int KCP   = 258;   // +2 dwords pad per row (TDM pad) -> no LDS bank conflicts

#if __has_builtin(__builtin_amdgcn_tensor_load_to_lds)
#define HAVE_TDM 1
#else
#define HAVE_TDM 0
#endif

// ---------------------------------------------------------------- WMMA helpers
__device__ inline v8f wmma4(v2f a, v2f b, v8f c) {
  // D = A(16x4) * B(4x16) + C, fp32. 8-arg form: (neg_a,A,neg_b,B,c_mod,C,reuse_a,reuse_b)
  return __builtin_amdgcn_wmma_f32_16x16x4_f32(false, a, false, b, (short)0, c, false, false);
}

// A-operand (16x4, row-major source with leading dim ld): lane<16 -> K{0,1}, lane>=16 -> K{2,3}
__device__ inline v2f load_a(const float* P, int ld, int row0, int k0) {
  int lane = threadIdx.x & 31;
  const float* q = P + (size_t)(row0 + (lane & 15)) * ld + k0 + ((lane >> 4) << 1);
  v2f r; r.x = q[0]; r.y = q[1];
  return r;
}

// B-operand where B[k][n] = W[n][k] (W row-major MxK, used as transposed right matrix)
__device__ inline v2f load_bT(const float* W, int ld, int n0, int k0) {
  int lane = threadIdx.x & 31;
  const float* q = W + (size_t)(n0 + (lane & 15)) * ld + k0 + ((lane >> 4) << 1);
  v2f r; r.x = q[0]; r.y = q[1];
  return r;
}

// B-operand, row-major KxN source: B[k][n] direct
__device__ inline v2f load_bk(const float* Bm, int ld, int k0, int n0) {
  int lane = threadIdx.x & 31;
  int k = k0 + ((lane >> 4) << 1);
  const float* q = Bm + (size_t)k * ld + n0 + (lane & 15);
  v2f r; r.x = q[0]; r.y = q[ld];
  return r;
}

// C/D 16x16 fp32 store: VGPR j, lane<16 -> (M=j,N=lane); lane>=16 -> (M=8+j,N=lane-16)
__device__ inline void store_c(float* D, int ld, int m0, int n0, v8f c) {
  int lane = threadIdx.x & 31;
  int n  = n0 + (lane & 15);
  int mb = m0 + ((lane >> 4) << 3);
#pragma unroll
  for (int j = 0; j < 8; ++j) D[(size_t)(mb + j) * ld + n] = c[j];
}

// ------------------------------------------------- TDM: 2D tile global -> LDS
// Loads tile1 rows x tile0 fp32 elements, row stride stride0 (elements), into
// LDS at ldsp with KCP-dword row pitch (TDM pad: 2 dwords after every 256).
__device__ inline void tdm_load_2d(const void* ldsp, const float* gptr,
                                   unsigned tile0, unsigned tile1,
                                   unsigned long long stride0) {
#if HAVE_TDM
  unsigned lds_off = (unsigned)(size_t)ldsp;           // low 32 bits = LDS byte offset
  unsigned long long ga = (unsigned long long)(const void*)gptr;
  v4u g0;
  g0[0] = 1u;                                          // count=1, user descriptor
  g0[1] = lds_off;                                     // lds_addr
  g0[2] = (unsigned)ga;                                // global_addr[31:0]
  g0[3] = (unsigned)((ga >> 32) & 0x01FFFFFFu) | (2u << 30); // addr[56:32] | type=2
  unsigned d[8];
  d[0] = (2u << 16)            // data_size = 4B
       | (1u << 20)            // pad_enable
       | (7u << 22)            // pad_interval = 256 dwords
       | (1u << 25);           // pad_amount  = 2 dwords
  unsigned long long dim0 = 0x40000000ull, dim1 = 0x40000000ull; // huge: no OOB clamp
  d[1] = (unsigned)((dim0 & 0xFFFFull) << 16);
  d[2] = (unsigned)((dim0 >> 16) & 0xFFFFull) | (unsigned)((dim1 & 0xFFFFull) << 16);
  d[3] = (unsigned)((dim1 >> 16) & 0xFFFFull) | (tile0 << 16);   // tile_dim0
  d[4] = tile1 & 0xFFFFu;                                        // tile_dim1
  d[5] = (unsigned)(stride0 & 0xFFFFFFFFull);                    // tensor_dim0_stride lo
  d[6] = (unsigned)((stride0 >> 32) & 0xFFFFull);                // stride hi | dim1_stride lo(0)
  d[7] = 0u;
  v8i g1;
#pragma unroll
  for (int i = 0; i < 8; ++i) g1[i] = (int)d[i];
  v4i z4 = {};
  v8i z8 = {};
  __builtin_amdgcn_tensor_load_to_lds(g0, g1, z4, z4, z8, 0);
#else
  int lane = threadIdx.x & 31;
  float* L = (float*)ldsp;
  for (unsigned i = lane; i < tile0 * tile1; i += 32) {
    unsigned r = i / tile0, c = i % tile0;
    L[r * KCP + c] = gptr[(size_t)r * stride0 + c];
  }
#endif
}

__device__ inline void tdm_wait() {
#if HAVE_TDM
  __builtin_amdgcn_s_wait_tensorcnt(0);
#endif
}

// ---------------------------------------------------------------- tiny kernels
__global__ void k_scal(const float* lr, const float* le, float* scal) {
  if (threadIdx.x == 0) {
    float rho = expf(lr[0]);
    scal[0] = rho;
    scal[1] = expf(le[0]);
    scal[2] = 1.0f / (rho + 1e-12f);
  }
}

__global__ void k_init(const float* y, const float* u_in, float* z, float* u, int n) {
  int i = blockIdx.x * blockDim.x + threadIdx.x;
  if (i < n) { z[i] = y[i]; u[i] = u_in[i]; }
}

__global__ void k_w(const float* z, const float* u, float* w, int n) {
  int i = blockIdx.x * blockDim.x + threadIdx.x;
  if (i < n) w[i] = z[i] - u[i];
}

// ------------------------------------------- Gram: S_block = [[Sr,-Si],[Si,Sr]]
// Sr = I/(rho+eps) + Ar*Ar^T + Ai*Ai^T ; Si = Ai*Ar^T - Ar*Ai^T   (64x64 each)
__global__ __launch_bounds__(256) void k_gram(const float* __restrict__ A,
                                              const float* __restrict__ scal,
                                              float* __restrict__ Sblk) {
  int t  = blockIdx.x * 8 + (threadIdx.x >> 5);
  int m0 = (t >> 2) * 16, n0 = (t & 3) * 16;
  int lane = threadIdx.x & 31;
  const float* Ar = A;
  const float* Ai = A + (size_t)MDIM * NDIM;
  v8f accR = {}, accI = {};
  for (int k = 0; k < NDIM; k += 4) {
    v2f aAr = load_a(Ar, NDIM, m0, k);
    v2f aAi = load_a(Ai, NDIM, m0, k);
    v2f bAr = load_bT(Ar, NDIM, n0, k);
    v2f bAi = load_bT(Ai, NDIM, n0, k);
    accR = wmma4(aAr, bAr, accR);
    accR = wmma4(aAi, bAi, accR);
    accI = wmma4(aAi, bAr, accI);
    v2f nb = -bAi;
    accI = wmma4(aAr, nb, accI);
  }
  float invr = scal[2];
  int n  = n0 + (lane & 15);
  int mb = m0 + ((lane >> 4) << 3);
#pragma unroll
  for (int j = 0; j < 8; ++j) {
    int m = mb + j;
    float sr = accR[j] + (m == n ? invr : 0.0f);
    float si = accI[j];
    Sblk[m * 128 + n]              = sr;
    Sblk[m * 128 + n + 64]         = -si;
    Sblk[(m + 64) * 128 + n]       = si;
    Sblk[(m + 64) * 128 + n + 64]  = sr;
  }
}

// ------------------------------------ single-block Gauss-Jordan: Sinv = Sblk^-1
__global__ __launch_bounds__(256) void k_inv(float* Sblk, float* Sinv) {
  __shared__ float colp[128];
  int tid = threadIdx.x;
  for (int i = tid; i < 128 * 128; i += 256)
    Sinv[i] = ((i >> 7) == (i & 127)) ? 1.0f : 0.0f;
  __syncthreads();
  for (int p = 0; p < 128; ++p) {
    if (tid < 128) colp[tid] = Sblk[tid * 128 + p];
    __syncthreads();
    float piv = 1.0f / Sblk[p * 128 + p];
    __syncthreads();
    if (tid < 128) { Sblk[p * 128 + tid] *= piv; Sinv[p * 128 + tid] *= piv; }
    __syncthreads();
    for (int i = tid; i < 128 * 128; i += 256) {
      int r = i >> 7, j = i & 127;
      if (r != p) {
        float f = colp[r];
        Sblk[i] -= f * Sblk[p * 128 + j];
        Sinv[i] -= f * Sinv[p * 128 + j];
      }
    }
    __syncthreads();
  }
}

// --------------------------- b = r_n + rho * A^H(w),  w=(z-u) stored [wr|wi] B x 128
__global__ __launch_bounds__(256) void k_bmat(const float* __restrict__ r_n,
                                              const float* __restrict__ w,
                                              const float* __restrict__ A,
                                              const float* __restrict__ scal,
                                              float* __restrict__ b) {
  int bi0 = blockIdx.x * 16;
  int wv = threadIdx.x >> 5, lane = threadIdx.x & 31;
  int n0 = (blockIdx.y * 8 + wv) * 16;
  const float* Ar = A;
  const float* Ai = A + (size_t)MDIM * NDIM;
  v8f accR = {}, accI = {};
#pragma unroll 4
  for (int k = 0; k < MDIM; k += 4) {
    v2f awr = load_a(w,       128, bi0, k);
    v2f awi = load_a(w + 64,  128, bi0, k);
    v2f bAr = load_bk(Ar, NDIM, k, n0);
    v2f bAi = load_bk(Ai, NDIM, k, n0);
    accR = wmma4(awr, bAr, accR);   // wr @ Ar
    accR = wmma4(awi, bAi, accR);   // + wi @ Ai
    accI = wmma4(awi, bAr, accI);   // wi @ Ar
    v2f nb = -bAi;
    accI = wmma4(awr, nb, accI);    // - wr @ Ai
  }
  float rho = scal[0];
  int n  = n0 + (lane & 15);
  int mb = bi0 + ((lane >> 4) << 3);
#pragma unroll
  for (int j = 0; j < 8; ++j) {
    size_t bi = (size_t)(mb + j);
    b[(bi * 2 + 0) * NDIM + n] = r_n[(bi * 2 + 0) * NDIM + n] + rho * accR[j];
    b[(bi * 2 + 1) * NDIM + n] = r_n[(bi * 2 + 1) * NDIM + n] + rho * accI[j];
  }
}

// --------------------------- R = [Ab_r | Ab_i] = A*b   (B x 128, K = 8192, TDM-staged)
__global__ __launch_bounds__(256) void k_cmm_Ab(const float* __restrict__ b,
                                                const float* __restrict__ A,
                                                float* __restrict__ R) {
  __shared__ float lds[2 * 16 * KCP];
  int bi0 = blockIdx.x * 16;
  int wv = threadIdx.x >> 5, lane = threadIdx.x & 31;
  int comp = wv >> 2;        // 0 -> Ab_r columns, 1 -> Ab_i columns
  int n0 = (wv & 3) * 16;    // output column tile within the 64
  const float* Ar = A;
  const float* Ai = A + (size_t)MDIM * NDIM;
  v8f acc = {};
  for (int kk = 0; kk < NDIM; kk += KC) {
    if (wv == 0) {
      tdm_load_2d(&lds[0],        b + ((size_t)bi0 * 2 + 0) * NDIM + kk, KC, 16, 2ull * NDIM);
      tdm_load_2d(&lds[16 * KCP], b + ((size_t)bi0 * 2 + 1) * NDIM + kk, KC, 16, 2ull * NDIM);
      tdm_wait();
    }
    __syncthreads();
    int kr = (lane >> 4) << 1;
#pragma unroll 4
    for (int kl = 0; kl < KC; kl += 4) {
      v2f ar = *(const v2f*)&lds[(lane & 15) * KCP + kl + kr];
      v2f ai = *(const v2f*)&lds[16 * KCP + (lane & 15) * KCP + kl + kr];
      int kg = kk + kl + kr;
      const float* pr = Ar + (size_t)(n0 + (lane & 15)) * NDIM + kg;
      const float* pi = Ai + (size_t)(n0 + (lane & 15)) * NDIM + kg;
      v2f bAr; bAr.x = pr[0]; bAr.y = pr[1];
      v2f bAi; bAi.x = pi[0]; bAi.y = pi[1];
      if (comp == 0) {
        acc = wmma4(ar, bAr, acc);      // b_r @ Ar^T
        v2f nb = -bAi;
        acc = wmma4(ai, nb, acc);       // - b_i @ Ai^T
      } else {
        acc = wmma4(ai, bAr, acc);      // b_i @ Ar^T
        acc = wmma4(ar, bAi, acc);      // + b_r @ Ai^T
      }
    }
    __syncthreads();
  }
  store_c(R, 128, bi0, comp * 64 + n0, acc);
}

// --------------------------- T = R @ Sinv^T  (B x 128, K = 128)
__global__ __launch_bounds__(256) void k_solve(const float* __restrict__ R,
                                               const float* __restrict__ Sinv,
                                               float* __restrict__ T) {
  int bi0 = blockIdx.x * 16;
  int wv = threadIdx.x >> 5;
  int n0 = wv * 16;
  v8f acc = {};
#pragma unroll 4
  for (int k = 0; k < 128; k += 4) {
    v2f a  = load_a(R, 128, bi0, k);
    v2f bb = load_bT(Sinv, 128, n0, k);
    acc = wmma4(a, bb, acc);
  }
  store_c(T, 128, bi0, n0, acc);
}

// --------------------------- x = relu(b_r - [T_r,T_i]@[Ar;Ai]) ; x_i = 0
__global__ __launch_bounds__(256) void k_xupd(const float* __restrict__ b,
                                              const float* __restrict__ T,
                                              const float* __restrict__ A,
                                              float* __restrict__ xout) {
  int bi0 = blockIdx.x * 16;
  int wv = threadIdx.x >> 5, lane = threadIdx.x & 31;
  int n0 = (blockIdx.y * 8 + wv) * 16;
  const float* Ar = A;
  const float* Ai = A + (size_t)MDIM * NDIM;
  v8f acc = {};
#pragma unroll 4
  for (int k = 0; k < 128; k += 4) {
    v2f a = load_a(T, 128, bi0, k);
    const float* Bm = (k < 64) ? (Ar + (size_t)k * NDIM) : (Ai + (size_t)(k - 64) * NDIM);
    int kk = (lane >> 4) << 1;
    const float* q = Bm + (size_t)kk * NDIM + n0 + (lane & 15);
    v2f bb; bb.x = q[0]; bb.y = q[NDIM];
    acc = wmma4(a, bb, acc);
  }
  int n  = n0 + (lane & 15);
  int mb = bi0 + ((lane >> 4) << 3);
#pragma unroll
  for (int j = 0; j < 8; ++j) {
    size_t bi = (size_t)(mb + j);
    float xr = b[(bi * 2 + 0) * NDIM + n] - acc[j];
    xout[(bi * 2 + 0) * NDIM + n] = xr > 0.0f ? xr : 0.0f;
    xout[(bi * 2 + 1) * NDIM + n] = 0.0f;
  }
}

// --------------------------- AxR = [x_r@Ar^T | x_r@Ai^T]  (B x 128, K=8192, TDM-staged)
__global__ __launch_bounds__(256) void k_cmm_Ax(const float* __restrict__ x,
                                                const float* __restrict__ A,
                                                float* __restrict__ AxR) {
  __shared__ float lds[16 * KCP];
  int bi0 = blockIdx.x * 16;
  int wv = threadIdx.x >> 5, lane = threadIdx.x & 31;
  int comp = wv >> 2;
  int n0 = (wv & 3) * 16;
  const float* W = A + (size_t)comp * MDIM * NDIM;   // Ar or Ai
  v8f acc = {};
  for (int kk = 0; kk < NDIM; kk += KC) {
    if (wv == 0) {
      tdm_load_2d(&lds[0], x + (size_t)bi0 * 2 * NDIM + kk, KC, 16, 2ull * NDIM);
      tdm_wait();
    }
    __syncthreads();
    int kr = (lane >> 4) << 1;
#pragma unroll 4
    for (int kl = 0; kl < KC; kl += 4) {
      v2f a = *(const v2f*)&lds[(lane & 15) * KCP + kl + kr];
      const float* q = W + (size_t)(n0 + (lane & 15)) * NDIM + kk + kl + kr;
      v2f bb; bb.x = q[0]; bb.y = q[1];
      acc = wmma4(a, bb, acc);
    }
    __syncthreads();
  }
  store_c(AxR, 128, bi0, comp * 64 + n0, acc);
}

// --------------------------- prox + dual update (per batch)
__global__ __launch_bounds__(128) void k_prox(const float* __restrict__ AxR,
                                              const float* __restrict__ y,
                                              float* __restrict__ z,
                                              float* __restrict__ u,
                                              const float* __restrict__ scal,
                                              float* u_out) {
  __shared__ float red[128];
  int b = blockIdx.x, t = threadIdx.x;
  float ax = AxR[b * 128 + t];
  float uu = u[b * 128 + t];
  float yy = y[b * 128 + t];
  float v = ax + uu - yy;
  red[t] = v * v;
  __syncthreads();
  for (int s = 64; s > 0; s >>= 1) {
    if (t < s) red[t] += red[t + s];
    __syncthreads();
  }
  float nrm = sqrtf(red[0]);
  float sc = fminf(1.0f, scal[1] / (nrm + 1e-12f));
  float zz = yy + v * sc;
  float un = uu + ax - zz;
  z[b * 128 + t] = zz;
  u[b * 128 + t] = un;
  if (u_out) u_out[b * 128 + t] = un;
}

// ------------------------------------------------------------------- dispatch
extern "C" void kernel_launch(void* const* d_in, const int* in_sizes, int n_in,
                              void* d_out, int out_size, void* d_ws, size_t ws_size,
                              hipStream_t stream) {
  const float* r_n  = (const float*)d_in[0];
  const float* y    = (const float*)d_in[1];
  const float* u_in = (const float*)d_in[2];
  const float* A    = (const float*)d_in[3];
  const float* lr   = (const float*)d_in[4];
  const float* le   = (const float*)d_in[5];
  float* out = (float*)d_out;
  float* ws  = (float*)d_ws;

  float* b    = ws;                                     // 1024*2*8192
  float* R    = b    + (size_t)BATCH * 2 * NDIM;        // 1024*128
  float* T    = R    + (size_t)BATCH * 128;
  float* Ax   = T    + (size_t)BATCH * 128;
  float* z    = Ax   + (size_t)BATCH * 128;
  float* u    = z    + (size_t)BATCH * 128;
  float* w    = u    + (size_t)BATCH * 128;
  float* Sblk = w    + (size_t)BATCH * 128;             // 128*128
  float* Sinv = Sblk + 128 * 128;
  float* scal = Sinv + 128 * 128;

  float* xout = out;
  float* uout = out + (size_t)BATCH * 2 * NDIM;

  k_scal<<<1, 32, 0, stream>>>(lr, le, scal);
  k_init<<<512, 256, 0, stream>>>(y, u_in, z, u, BATCH * 128);
  k_gram<<<2, 256, 0, stream>>>(A, scal, Sblk);
  k_inv<<<1, 256, 0, stream>>>(Sblk, Sinv);

  for (int s = 0; s < 3; ++s) {
    k_w<<<512, 256, 0, stream>>>(z, u, w, BATCH * 128);
    k_bmat<<<dim3(64, 64), 256, 0, stream>>>(r_n, w, A, scal, b);
    k_cmm_Ab<<<64, 256, 0, stream>>>(b, A, R);
    k_solve<<<64, 256, 0, stream>>>(R, Sinv, T);
    k_xupd<<<dim3(64, 64), 256, 0, stream>>>(b, T, A, xout);
    k_cmm_Ax<<<64, 256, 0, stream>>>(xout, A, Ax);
    k_prox<<<1024, 128, 0, stream>>>(Ax, y, z, u, scal, (s == 2) ? uout : nullptr);
  }
}